// Net_14525579395835
// MI455X (gfx1250) — compile-verified
//
#include <hip/hip_runtime.h>

typedef __attribute__((ext_vector_type(2))) float v2f;
typedef __attribute__((ext_vector_type(8))) float v8f;
typedef int v4i __attribute__((vector_size(16)));   // b128 payload type

#define NN  50000
#define NNP 50048      // NN padded to a multiple of 64
#define NE  800000

// ---------------------------------------------------------------------------
// CDNA5 async global->LDS staging (ASYNCcnt path), with sync fallback
// ---------------------------------------------------------------------------
#if __has_builtin(__builtin_amdgcn_global_load_async_to_lds_b128)
#define GCN_ASYNC 1
#else
#define GCN_ASYNC 0
#endif

#if GCN_ASYNC
#define GLD_ASYNC_B128(lds, gbl)                                   \
    __builtin_amdgcn_global_load_async_to_lds_b128(                \
        (__attribute__((address_space(1))) v4i*)(gbl),             \
        (__attribute__((address_space(3))) v4i*)(lds), 0, 0)

__device__ __forceinline__ void wait_async0()
{
#if __has_builtin(__builtin_amdgcn_s_wait_asynccnt)
    __builtin_amdgcn_s_wait_asynccnt(0);
#else
    asm volatile("s_wait_asynccnt 0" ::: "memory");
#endif
}
#endif

// ---------------------------------------------------------------------------
// 16-deep K slab of WMMAs on one staged tile pair
// ---------------------------------------------------------------------------
__device__ __forceinline__ void wmma_k16(
    const float (*__restrict__ AsT)[20], const float (*__restrict__ BsT)[72],
    int wrow, int wcol, int ml, int kp, v8f& acc0, v8f& acc1)
{
#pragma unroll
    for (int kk = 0; kk < 16; kk += 4) {
        // A 16x4 frag: lanes 0-15 K=kk+{0,1}; lanes 16-31 K=kk+{2,3}
        const v2f a = *(const v2f*)&AsT[wrow + ml][kk + kp];
        v2f b0, b1;
        b0.x = BsT[kk + kp + 0][wcol + ml];
        b0.y = BsT[kk + kp + 1][wcol + ml];
        b1.x = BsT[kk + kp + 0][wcol + 16 + ml];
        b1.y = BsT[kk + kp + 1][wcol + 16 + ml];
        acc0 = __builtin_amdgcn_wmma_f32_16x16x4_f32(
            false, a, false, b0, (short)0, acc0, false, false);
        acc1 = __builtin_amdgcn_wmma_f32_16x16x4_f32(
            false, a, false, b1, (short)0, acc1, false, false);
    }
}

// ---------------------------------------------------------------------------
// WMMA fp32 GEMM:  C[M,F] = A[M,K] @ B[K,F],  all dims pre-padded:
//   M = NNP (mult of 64), K mult of 16, F mult of 64.
// Block = 256 threads (8 waves), tile BM=64 BN=64 BK=16.
// Async path: double-buffered LDS, one barrier per K-tile, DMA writes LDS
// directly (no VGPR round-trip). Fallback: two-barrier sync staging.
// LDS strides 20/72 are bank-conflict-free and keep b128 stores aligned.
// ---------------------------------------------------------------------------
__global__ __launch_bounds__(256) void gemm_wmma_f32(
    const float* __restrict__ A, const float* __restrict__ B,
    float* __restrict__ C, int K, int F)
{
#if GCN_ASYNC
    __shared__ float As[2][64][20];
    __shared__ float Bs[2][16][72];
#else
    __shared__ float As[1][64][20];
    __shared__ float Bs[1][16][72];
#endif

    const int tid  = threadIdx.x;
    const int lane = tid & 31;
    const int wave = tid >> 5;
    const int wrow = (wave & 3) << 4;    // 0,16,32,48
    const int wcol = (wave >> 2) << 5;   // 0,32
    const int row0 = blockIdx.x << 6;
    const int col0 = blockIdx.y << 6;
    const int ml   = lane & 15;
    const int kp   = (lane >> 4) << 1;

    // staging indices (one float4 / b128 per thread per tile)
    const int ar = tid >> 2;             // 0..63
    const int ac = (tid & 3) << 2;       // 0,4,8,12
    const int br = tid >> 4;             // 0..15
    const int bc = (tid & 15) << 2;      // 0..60

    const float*  aptr  = A + (size_t)(row0 + ar) * K + ac;
    const float*  bptr  = B + (size_t)br * F + (col0 + bc);
    const size_t  bstep = (size_t)16 * F;

    v8f acc0 = {};
    v8f acc1 = {};

#if GCN_ASYNC
    // prologue: DMA tile 0 into buffer 0
    GLD_ASYNC_B128(&As[0][ar][ac], aptr);
    GLD_ASYNC_B128(&Bs[0][br][bc], bptr);
    aptr += 16;
    bptr += bstep;

    int buf = 0;
    for (int k0 = 0; k0 < K; k0 += 16) {
        wait_async0();                   // my tile-t copy complete
        __syncthreads();                 // all copies visible; prev tile consumed
        const int cur = buf;
        if (k0 + 16 < K) {               // DMA tile t+1 while computing tile t
            buf ^= 1;
            GLD_ASYNC_B128(&As[buf][ar][ac], aptr);
            GLD_ASYNC_B128(&Bs[buf][br][bc], bptr);
            aptr += 16;
            bptr += bstep;
        }
        wmma_k16(As[cur], Bs[cur], wrow, wcol, ml, kp, acc0, acc1);
    }
#else
    for (int k0 = 0; k0 < K; k0 += 16) {
        const float4 av = *(const float4*)aptr;
        const float4 bv = *(const float4*)bptr;
        const bool more = (k0 + 16) < K;
        aptr = more ? (aptr + 16) : aptr;
        bptr = more ? (bptr + bstep) : bptr;

        __syncthreads();
        *(float4*)&As[0][ar][ac] = av;
        *(float4*)&Bs[0][br][bc] = bv;
        __syncthreads();

        __builtin_prefetch(aptr, 0, 3);
        __builtin_prefetch(bptr, 0, 3);

        wmma_k16(As[0], Bs[0], wrow, wcol, ml, kp, acc0, acc1);
    }
#endif

    // C/D layout: VGPR v -> M = v (lanes 0-15) / 8+v (lanes 16-31); N = lane&15
    const int n0 = col0 + wcol + ml;
    const int m0 = row0 + wrow + ((lane >> 4) << 3);
    float* cptr = C + (size_t)m0 * F + n0;
#pragma unroll
    for (int v = 0; v < 8; ++v) {
        cptr[0]  = acc0[v];
        cptr[16] = acc1[v];
        cptr += F;
    }
}

// ---------------------------------------------------------------------------
// Padding copies (once per call; tiny vs. 23.3 TB/s)
// ---------------------------------------------------------------------------
__global__ __launch_bounds__(256) void k_pad_x(const float* __restrict__ x,
                                               float* __restrict__ xp)
{
    const int r = blockIdx.x;                  // 0..NNP-1
    const bool real = r < NN;
    const float* __restrict__ s = x + (size_t)r * 767;
    float* __restrict__ d = xp + (size_t)r * 768;
    for (int c = threadIdx.x; c < 768; c += 256)
        d[c] = (real && c < 767) ? s[c] : 0.0f;
}

__global__ __launch_bounds__(256) void k_pad_w1(const float* __restrict__ W,
                                                float* __restrict__ Wp)
{
    const int r = blockIdx.x;                  // 0..767
    for (int c = threadIdx.x; c < 640; c += 256)
        Wp[(size_t)r * 640 + c] = (r < 767) ? W[(size_t)r * 640 + c] : 0.0f;
}

__global__ __launch_bounds__(64) void k_pad_w6(const float* __restrict__ W,
                                               float* __restrict__ Wp)
{
    const int r = blockIdx.x;                  // 0..127
    const int c = threadIdx.x;                 // 0..63
    Wp[(size_t)r * 64 + c] = (c < 10) ? W[(size_t)r * 10 + c] : 0.0f;
}

// ---------------------------------------------------------------------------
// Degree / normalization
// ---------------------------------------------------------------------------
__global__ __launch_bounds__(256) void k_fill(float* p, float v, int n)
{
    int i = blockIdx.x * blockDim.x + threadIdx.x;
    if (i < n) p[i] = v;
}

__global__ __launch_bounds__(256) void k_deg(const int* __restrict__ dst,
                                             float* __restrict__ deg, int e)
{
    int i = blockIdx.x * blockDim.x + threadIdx.x;
    if (i < e) atomicAdd(&deg[dst[i]], 1.0f);
}

__global__ __launch_bounds__(256) void k_rsqrt(float* p, int n)
{
    int i = blockIdx.x * blockDim.x + threadIdx.x;
    if (i < n) p[i] = rsqrtf(p[i]);
}

// ---------------------------------------------------------------------------
// Message passing
// ---------------------------------------------------------------------------
// Acc[r,:] = dis[r]^2 * T[r,:]   (self-loop term initializes the accumulator)
__global__ __launch_bounds__(64) void k_selfloop(const float* __restrict__ T,
                                                 const float* __restrict__ dis,
                                                 float* __restrict__ Acc, int Fs)
{
    const int r = blockIdx.x;
    const float d = dis[r];
    const float w = d * d;
    const float4* __restrict__ t = (const float4*)(T + (size_t)r * Fs);
    float4* __restrict__ a = (float4*)(Acc + (size_t)r * Fs);
    const int n4 = Fs >> 2;
    for (int c = threadIdx.x; c < n4; c += 64) {
        float4 v = t[c];
        v.x *= w; v.y *= w; v.z *= w; v.w *= w;
        a[c] = v;
    }
}

// one block per edge: Acc[dst,:] += dis[src]*dis[dst] * T[src,:]
__global__ __launch_bounds__(64) void k_scatter(const int* __restrict__ src,
                                                const int* __restrict__ dst,
                                                const float* __restrict__ dis,
                                                const float* __restrict__ T,
                                                float* __restrict__ Acc, int Fs)
{
    const int e = blockIdx.x;
    const int s = src[e];
    const int d = dst[e];
    const float w = dis[s] * dis[d];
    const float4* __restrict__ t = (const float4*)(T + (size_t)s * Fs);
    float* __restrict__ a = Acc + (size_t)d * Fs;
    const int n4 = Fs >> 2;
    for (int c = threadIdx.x; c < n4; c += 64) {
        const float4 v = t[c];
        const int f = c << 2;
        atomicAdd(&a[f + 0], w * v.x);
        atomicAdd(&a[f + 1], w * v.y);
        atomicAdd(&a[f + 2], w * v.z);
        atomicAdd(&a[f + 3], w * v.w);
    }
}

__global__ __launch_bounds__(64) void k_bias_relu(float* __restrict__ H,
                                                  const float* __restrict__ b,
                                                  int Fs)
{
    const int r = blockIdx.x;
    float4* __restrict__ h = (float4*)(H + (size_t)r * Fs);
    const float4* __restrict__ bb = (const float4*)b;
    const int n4 = Fs >> 2;
    for (int c = threadIdx.x; c < n4; c += 64) {
        float4 v = h[c];
        const float4 vb = bb[c];
        v.x = fmaxf(v.x + vb.x, 0.0f);
        v.y = fmaxf(v.y + vb.y, 0.0f);
        v.z = fmaxf(v.z + vb.z, 0.0f);
        v.w = fmaxf(v.w + vb.w, 0.0f);
        h[c] = v;
    }
}

// last layer: bias + log_softmax over 10 classes, accumulator stride 64
__global__ __launch_bounds__(256) void k_bias_logsoftmax10(
    const float* __restrict__ Acc, const float* __restrict__ b,
    float* __restrict__ out, int n)
{
    const int row = blockIdx.x * blockDim.x + threadIdx.x;
    if (row >= n) return;
    float v[10];
    float mx = -3.4e38f;
#pragma unroll
    for (int f = 0; f < 10; ++f) {
        v[f] = Acc[(size_t)row * 64 + f] + b[f];
        mx = fmaxf(mx, v[f]);
    }
    float s = 0.0f;
#pragma unroll
    for (int f = 0; f < 10; ++f) s += __expf(v[f] - mx);
    const float lse = mx + __logf(s);
#pragma unroll
    for (int f = 0; f < 10; ++f) out[(size_t)row * 10 + f] = v[f] - lse;
}

// ---------------------------------------------------------------------------
// Launch
// ---------------------------------------------------------------------------
extern "C" void kernel_launch(void* const* d_in, const int* in_sizes, int n_in,
                              void* d_out, int out_size, void* d_ws, size_t ws_size,
                              hipStream_t stream)
{
    (void)in_sizes; (void)n_in; (void)out_size; (void)ws_size;

    const float* x   = (const float*)d_in[0];
    const int*   ei  = (const int*)d_in[1];
    const int*   src = ei;        // edge_index[0]
    const int*   dst = ei + NE;   // edge_index[1]

    const float* bl[6] = { (const float*)d_in[3],  (const float*)d_in[5],
                           (const float*)d_in[7],  (const float*)d_in[9],
                           (const float*)d_in[11], (const float*)d_in[13] };

    // workspace (floats): dis[NNP] | W1p[768*640] | W6p[128*64]
    //                     | bufA[NNP*768] (xpad, then activations) | bufB[NNP*640]
    float* dis  = (float*)d_ws;
    float* W1p  = dis + NNP;
    float* W6p  = W1p + 768 * 640;
    float* bufA = W6p + 128 * 64;
    float* bufB = bufA + (size_t)NNP * 768;

    // deg (incl. self-loop) -> dis = rsqrt(deg)
    k_fill <<<(NN + 255) / 256, 256, 0, stream>>>(dis, 1.0f, NN);
    k_deg  <<<(NE + 255) / 256, 256, 0, stream>>>(dst, dis, NE);
    k_rsqrt<<<(NN + 255) / 256, 256, 0, stream>>>(dis, NN);

    // padded operands
    k_pad_x <<<NNP, 256, 0, stream>>>(x, bufA);
    k_pad_w1<<<768, 256, 0, stream>>>((const float*)d_in[2],  W1p);
    k_pad_w6<<<128, 64,  0, stream>>>((const float*)d_in[12], W6p);

    const int    Ks[6] = {768, 640, 512, 384, 256, 128};
    const int    Fs[6] = {640, 512, 384, 256, 128, 64};
    const float* Wp[6] = { W1p, (const float*)d_in[4], (const float*)d_in[6],
                           (const float*)d_in[8], (const float*)d_in[10], W6p };

    for (int l = 0; l < 6; ++l) {
        const int K = Ks[l];
        const int F = Fs[l];

        dim3 g(NNP / 64, F / 64);
        gemm_wmma_f32<<<g, 256, 0, stream>>>(bufA, Wp[l], bufB, K, F);

        k_selfloop<<<NN, 64, 0, stream>>>(bufB, dis, bufA, F);
        k_scatter <<<NE, 64, 0, stream>>>(src, dst, dis, bufB, bufA, F);

        if (l < 5) {
            k_bias_relu<<<NN, 64, 0, stream>>>(bufA, bl[l], F);
            // zero the 48 pad rows (stride F) so next layer's GEMM reads zeros
            k_fill<<<(48 * F + 255) / 256, 256, 0, stream>>>(
                bufA + (size_t)NN * F, 0.0f, 48 * F);
        } else {
            k_bias_logsoftmax10<<<(NN + 255) / 256, 256, 0, stream>>>(
                bufA, bl[5], (float*)d_out, NN);
        }
    }
}